// SubspaceLinopFactory_82025285419231
// MI455X (gfx1250) — compile-verified
//
#include <hip/hip_runtime.h>
#include <hip/hip_bf16.h>

// Problem constants (from reference): N=2, A=5, C=12, H=W=256, grid 512x512.
#define NB   2
#define AA   5
#define CC   12
#define HH   256
#define GG   512
#define PS   (GG*GG)          // 262144 complex per plane
#define NPL  (NB*CC*AA)       // 120 planes

typedef float v2f __attribute__((ext_vector_type(2)));
typedef float v8f __attribute__((ext_vector_type(8)));

__device__ __forceinline__ float2 cmulf(float2 a, float2 b) {
    return make_float2(a.x*b.x - a.y*b.y, a.x*b.y + a.y*b.x);
}

// ---------------------------------------------------------------------------
// CDNA5 async global<->LDS copies (ASYNCcnt path), per CDNA5 ISA.
// LDS address operand is a byte offset within the wave's LDS allocation.
// ---------------------------------------------------------------------------
__device__ __forceinline__ unsigned ldsOff(const void* p) {
    return (unsigned)(unsigned long long)
           (__attribute__((address_space(3))) const char*)p;
}
__device__ __forceinline__ void asyncLdB64(void* lds, const void* g) {
    asm volatile("global_load_async_to_lds_b64 %0, %1, off"
                 :: "v"(ldsOff(lds)), "v"(g) : "memory");
}
__device__ __forceinline__ void asyncStB64(void* g, const void* lds) {
    asm volatile("global_store_async_from_lds_b64 %0, %1, off"
                 :: "v"(g), "v"(ldsOff(lds)) : "memory");
}
__device__ __forceinline__ void asyncWait0() {
    asm volatile("s_wait_asynccnt 0x0" ::: "memory");
}

// ---------------------------------------------------------------------------
// Unified twiddle table: tw[t] = (cos(2*pi*t/512), DIR*sin(2*pi*t/512)).
// Every twiddle used by the 512-pt transform is a power of W_512:
//   16-pt DFT matrix entry  -> tw[((m*k)&15)*32]
//   inter-stage W_512^(k1*n2) -> tw[k1*n2]          (k1*n2 <= 465)
//   32-pt DIF stage W_len^j  -> tw[j*(512/len)]
// ---------------------------------------------------------------------------
template<int DIR, int NT>
__device__ __forceinline__ void buildTw(float2* tw, int tid)
{
#pragma unroll
    for (int t = tid; t < 512; t += NT) {
        float sn, csn;
        __sincosf((6.283185307179586f / 512.0f) * (float)t, &sn, &csn);
        tw[t] = make_float2(csn, (float)DIR * sn);
    }
}

// ---------------------------------------------------------------------------
// 512-point complex FFT of one line held in LDS (natural order), one wave32.
// Four-step: 512 = 16 x 32.  n = n2 + 32*n1,  X[k1 + 16*k2].
//   1) 16-pt DFT over n1 via V_WMMA_F32_16X16X4_F32 (complex 16x16 x 16x32,
//      4 K-steps x 2 N-tiles x 4 real wmma = 32 wmma / line)
//   2) twiddle W_512^(k1*n2), written back in place (disjoint residues)
//   3) 32-pt radix-2 DIF over n2 in LDS (bit-reversed out, fixed on store)
// Direction is folded into the tw table.  scr = LDS line (stride cs);
// dst may be global or alias scr (in-place column FFT).
// ---------------------------------------------------------------------------
__device__ __forceinline__ void fft512_line(float2* scr, int cs,
                                            float2* dst, int ds_,
                                            const float2* tw, int lane)
{
    // A fragments: 16x16 DFT matrix in 4 K-chunks of 16x4.
    // 16x4 f32 A layout: lanes 0-15 -> M=lane, K={0,1}; lanes 16-31 -> K={2,3}.
    int m  = lane & 15;
    int kh = (lane >> 4) << 1;
    v2f Ac[4], Asd[4], Asn[4];
#pragma unroll
    for (int s = 0; s < 4; ++s) {
#pragma unroll
        for (int j = 0; j < 2; ++j) {
            int    kk = 4*s + kh + j;
            float2 w  = tw[((m * kk) & 15) << 5];
            Ac [s][j] = w.x;     // Re(F16)
            Asd[s][j] = w.y;     // Im(F16)  (= DIR*sin)
            Asn[s][j] = -w.y;    // -Im(F16)
        }
    }

    // stage 1+2: WMMA 16-pt DFT + twiddle -> scr[k1*32 + n2]
#pragma unroll
    for (int t = 0; t < 2; ++t) {          // N-tiles: n2 in [16t, 16t+16)
        // Batch-load all B fragments for this tile (ds_load, overlapped).
        v2f br[4], bi[4];
#pragma unroll
        for (int s = 0; s < 4; ++s) {
#pragma unroll
            for (int j = 0; j < 2; ++j) {
                // 4x16 f32 B layout: vgpr j, lanes0-15 row 2j, lanes16-31 row 2j+1
                int n1 = 4*s + 2*j + (lane >> 4);
                int n2 = 16*t + (lane & 15);
                float2 xv = scr[(size_t)(n2 + 32*n1) * cs];
                br[s][j] = xv.x;
                bi[s][j] = xv.y;
            }
        }
        v8f cr = {0.f,0.f,0.f,0.f,0.f,0.f,0.f,0.f};
        v8f ci = {0.f,0.f,0.f,0.f,0.f,0.f,0.f,0.f};
#pragma unroll
        for (int s = 0; s < 4; ++s) {
            // Yr = cos*Xr + (-dir*sin)*Xi ; Yi = (dir*sin)*Xr + cos*Xi
            cr = __builtin_amdgcn_wmma_f32_16x16x4_f32(false, Ac [s], false, br[s], (short)0, cr, false, false);
            cr = __builtin_amdgcn_wmma_f32_16x16x4_f32(false, Asn[s], false, bi[s], (short)0, cr, false, false);
            ci = __builtin_amdgcn_wmma_f32_16x16x4_f32(false, Asd[s], false, br[s], (short)0, ci, false, false);
            ci = __builtin_amdgcn_wmma_f32_16x16x4_f32(false, Ac [s], false, bi[s], (short)0, ci, false, false);
        }
        // C/D layout: vgpr r, lanes0-15 -> M=r, lanes16-31 -> M=r+8; N=lane&15
        int n2  = 16*t + (lane & 15);
        int mhi = (lane >> 4) * 8;
#pragma unroll
        for (int r = 0; r < 8; ++r) {
            int    k1 = r + mhi;
            float2 w  = tw[k1 * n2];
            float2 y  = make_float2(cr[r], ci[r]);
            scr[(size_t)(k1*32 + n2) * cs] = cmulf(y, w);
        }
    }
    __syncthreads();

    // stage 3: 32-pt radix-2 DIF per k1 row (16 rows x 16 butterflies / stage)
#pragma unroll
    for (int len = 32; len >= 2; len >>= 1) {
        int half = len >> 1;
        int step = 512 / len;
#pragma unroll
        for (int q = 0; q < 8; ++q) {
            int bf  = lane + 32*q;
            int k1  = bf >> 4;
            int t2  = bf & 15;
            int blk = (t2 / half) * len;
            int j   = t2 % half;
            int i0  = k1*32 + blk + j;
            float2 u = scr[(size_t)i0 * cs];
            float2 v = scr[(size_t)(i0 + half) * cs];
            float2 su = make_float2(u.x + v.x, u.y + v.y);
            float2 df = make_float2(u.x - v.x, u.y - v.y);
            float2 w  = tw[j * step];
            scr[(size_t)i0 * cs]          = su;
            scr[(size_t)(i0 + half) * cs] = cmulf(df, w);
        }
        __syncthreads();
    }

    // output: X[k1 + 16*k2] = scr[k1*32 + brev5(k2)], staged in registers so
    // dst may alias scr (in-place column FFT)
    float2 tmp[16];
#pragma unroll
    for (int i = 0; i < 16; ++i) {
        int k  = lane + 32*i;
        int k1 = k & 15;
        int k2 = k >> 4;
        int p  = k1*32 + (int)(__brev((unsigned)k2) >> 27);
        tmp[i] = scr[(size_t)p * cs];
    }
    __syncthreads();
#pragma unroll
    for (int i = 0; i < 16; ++i) {
        int k = lane + 32*i;
        dst[(size_t)k * ds_] = tmp[i];
    }
}

// ---------------------------------------------------------------------------
// K1: xm = x * mps, centered zero-pad 256 -> 512.  planes [n][c][a].
// ---------------------------------------------------------------------------
__global__ __launch_bounds__(256)
void k_pad_expand(const float* __restrict__ xr, const float* __restrict__ xi,
                  const float* __restrict__ mr, const float* __restrict__ mi,
                  float2* __restrict__ w0)
{
    size_t idx = (size_t)blockIdx.x * 256 + threadIdx.x;   // NPL*PS total
    int pix   = (int)(idx & (PS - 1));
    int plane = (int)(idx >> 18);
    int y = pix >> 9, x = pix & 511;
    int a  = plane % AA;
    int nc = plane / AA;
    int c  = nc % CC;
    int n  = nc / CC;

    float2 v = make_float2(0.f, 0.f);
    int yy = y - 128, xx = x - 128;
    if ((unsigned)yy < HH && (unsigned)xx < HH) {
        size_t xo = ((size_t)(n*AA + a) * HH + yy) * HH + xx;
        size_t mo = ((size_t)c * HH + yy) * HH + xx;
        float2 xv = make_float2(xr[xo], xi[xo]);
        float2 mv = make_float2(mr[mo], mi[mo]);
        v = cmulf(xv, mv);
    }
    w0[idx] = v;
}

// ---------------------------------------------------------------------------
// K2/K6: FFT along x (contiguous). One wave per 512-pt row, 8 waves/block.
// Rows stream into LDS via async global->LDS copies (overlapped with the
// twiddle-table build), FFT in place, coalesced register stores back.
// ---------------------------------------------------------------------------
template<int DIR>
__global__ __launch_bounds__(256)
void k_fft_rows(float2* __restrict__ data)
{
    __shared__ float2 scr[8][512];
    __shared__ float2 tw[512];
    int wave = threadIdx.x >> 5;
    int lane = threadIdx.x & 31;
    size_t line = (size_t)blockIdx.x * 8 + wave;   // NPL*512 lines
    float2* row = data + line * GG;
    float2* s   = scr[wave];

#pragma unroll
    for (int i = 0; i < 16; ++i) {
        int k = lane + 32*i;
        asyncLdB64(&s[k], &row[k]);
    }
    buildTw<DIR, 256>(tw, threadIdx.x);    // overlaps the async copies
    asyncWait0();
    __syncthreads();                       // publish tw (scr is wave-private)
    fft512_line(s, 1, row, 1, tw, lane);
}

// ---------------------------------------------------------------------------
// K3/K5: FFT along y. Block stages a 512x16-column tile in LDS (pitch 17 to
// spread banks) with async copies; 16 waves transform one column each fully
// in place in LDS; tile is written back with async LDS->global stores
// (S_ENDPGM's implicit wait-idle covers ASYNCcnt completion).
// ---------------------------------------------------------------------------
#define TPITCH 17
template<int DIR>
__global__ __launch_bounds__(512)
void k_fft_cols(float2* __restrict__ data)
{
    __shared__ float2 tile[GG * TPITCH];
    __shared__ float2 tw[512];
    int lane = threadIdx.x & 31;
    int wave = threadIdx.x >> 5;                   // 16 waves = 16 columns
    int img  = blockIdx.x >> 5;                    // 32 col-groups per image
    int cg   = blockIdx.x & 31;
    float2* base = data + (size_t)img * PS + (size_t)cg * 16;

    int ccol = threadIdx.x & 15;
    int rbas = threadIdx.x >> 4;                   // 0..31
#pragma unroll
    for (int it = 0; it < 16; ++it) {
        int r = rbas + 32*it;
        asyncLdB64(&tile[r*TPITCH + ccol], &base[(size_t)r * GG + ccol]);
    }
    buildTw<DIR, 512>(tw, threadIdx.x);    // overlaps the async copies
    asyncWait0();
    __syncthreads();

    fft512_line(&tile[wave], TPITCH, &tile[wave], TPITCH, tw, lane);
    __syncthreads();

#pragma unroll
    for (int it = 0; it < 16; ++it) {
        int r = rbas + 32*it;
        asyncStB64(&base[(size_t)r * GG + ccol], &tile[r*TPITCH + ccol]);
    }
}

// ---------------------------------------------------------------------------
// K4: per-frequency 5x5 complex mix over A, batched over nc=24, in place.
// Folds all normalization: ortho fwd (1/512) * ortho inv (1/512) * OVERSAMP^2.
// ---------------------------------------------------------------------------
__global__ __launch_bounds__(256)
void k_mix(float2* __restrict__ data,
           const float* __restrict__ kr, const float* __restrict__ ki)
{
    size_t p = (size_t)blockIdx.x * 256 + threadIdx.x;   // PS pixels
    float2 K[AA][AA];
#pragma unroll
    for (int b = 0; b < AA; ++b)
#pragma unroll
        for (int a = 0; a < AA; ++a) {
            size_t o = (size_t)(b*AA + a) * PS + p;
            K[b][a] = make_float2(kr[o], ki[o]);
        }
    const float scale = 4.0f / (float)PS;
    for (int nc = 0; nc < NB*CC; ++nc) {
        if (nc + 1 < NB*CC)
            __builtin_prefetch(&data[(size_t)((nc + 1)*AA) * PS + p], 0, 1);
        float2 F[AA];
#pragma unroll
        for (int a = 0; a < AA; ++a)
            F[a] = data[(size_t)(nc*AA + a) * PS + p];
#pragma unroll
        for (int b = 0; b < AA; ++b) {
            float2 acc = make_float2(0.f, 0.f);
#pragma unroll
            for (int a = 0; a < AA; ++a) {
                float2 t = cmulf(K[b][a], F[a]);
                acc.x += t.x; acc.y += t.y;
            }
            data[(size_t)(nc*AA + b) * PS + p] =
                make_float2(acc.x * scale, acc.y * scale);
        }
    }
}

// ---------------------------------------------------------------------------
// K7: crop center 512->256, multiply conj(mps), sum over coils.
// ---------------------------------------------------------------------------
__global__ __launch_bounds__(256)
void k_combine(const float2* __restrict__ data,
               const float* __restrict__ mr, const float* __restrict__ mi,
               float2* __restrict__ out)
{
    size_t idx = (size_t)blockIdx.x * 256 + threadIdx.x;  // NB*AA*HH*HH
    int x = (int)(idx & (HH - 1));
    int y = (int)((idx >> 8) & (HH - 1));
    int b = (int)((idx >> 16) % AA);
    int n = (int)(idx / ((size_t)AA * HH * HH));

    float2 acc = make_float2(0.f, 0.f);
    size_t pix = (size_t)(y + 128) * GG + (x + 128);
#pragma unroll
    for (int c = 0; c < CC; ++c) {
        float2 v = data[((size_t)((n*CC + c)*AA + b)) * PS + pix];
        size_t mo = ((size_t)c * HH + y) * HH + x;
        float mrv = mr[mo], miv = mi[mo];      // conj
        acc.x += v.x * mrv + v.y * miv;
        acc.y += v.y * mrv - v.x * miv;
    }
    out[idx] = acc;
}

// ---------------------------------------------------------------------------
extern "C" void kernel_launch(void* const* d_in, const int* in_sizes, int n_in,
                              void* d_out, int out_size, void* d_ws, size_t ws_size,
                              hipStream_t stream)
{
    const float* xr = (const float*)d_in[0];
    const float* xi = (const float*)d_in[1];
    const float* mr = (const float*)d_in[2];
    const float* mi = (const float*)d_in[3];
    const float* kr = (const float*)d_in[4];
    const float* ki = (const float*)d_in[5];
    float2* w0  = (float2*)d_ws;               // 120 * 512^2 complex = 252 MB
    float2* out = (float2*)d_out;

    const int nLines = NPL * GG;               // 61440 rows / cols per pass

    k_pad_expand<<<(NPL * PS) / 256, 256, 0, stream>>>(xr, xi, mr, mi, w0);

    k_fft_rows<-1><<<nLines / 8, 256, 0, stream>>>(w0);
    k_fft_cols<-1><<<NPL * (GG / 16), 512, 0, stream>>>(w0);

    k_mix<<<PS / 256, 256, 0, stream>>>(w0, kr, ki);

    k_fft_cols<+1><<<NPL * (GG / 16), 512, 0, stream>>>(w0);
    k_fft_rows<+1><<<nLines / 8, 256, 0, stream>>>(w0);

    k_combine<<<(NB * AA * HH * HH) / 256, 256, 0, stream>>>(w0, mr, mi, out);
}